// Decoder_91018946937564
// MI455X (gfx1250) — compile-verified
//
#include <hip/hip_runtime.h>
#include <hip/hip_bf16.h>

// Decoder step: GRU cell + Bahdanau attention over the full POI embedding
// table + output dense. All GEMMs run on V_WMMA_F32_16X16X4_F32 (full f32
// to match the f32 reference). The tanh-score stage (the true bottleneck:
// 64*5000*512 tanh) uses the CDNA5 hardware v_tanh_f32 TRANS op, which
// co-executes with the VALU fma stream.

#define B_   64
#define POI_ 5000
#define EMB_ 256
#define U_   512
#define QD_  256

typedef __attribute__((ext_vector_type(2))) float v2f;
typedef __attribute__((ext_vector_type(8))) float v8f;

#define LOG2E 1.4426950408889634f

__device__ __forceinline__ float fexp(float x)     { return __builtin_amdgcn_exp2f(x * LOG2E); }
__device__ __forceinline__ float fsigmoid(float x) { return __builtin_amdgcn_rcpf(1.0f + fexp(-x)); }
// CDNA5 hardware transcendental tanh (ISA lists TANH among TRANS ops).
__device__ __forceinline__ float ftanh_hw(float x) {
  float r;
  asm("v_tanh_f32 %0, %1" : "=v"(r) : "v"(x));
  return r;
}
__device__ __forceinline__ int   imin(int a, int b){ return a < b ? a : b; }

__device__ __forceinline__ v8f wmma4(v2f a, v2f b, v8f c) {
  // D = A(16x4 f32) * B(4x16 f32) + C(16x16 f32)
  return __builtin_amdgcn_wmma_f32_16x16x4_f32(false, a, false, b, (short)0, c,
                                               false, false);
}

// ---------------------------------------------------------------------------
// Fused GRU cell: x1 = concat(emb[x], query); 6 fused 16x16 WMMA tiles
// (xz,xr,xh,hz,hr,hh) per wave, K=512. Writes h_new everywhere it is needed.
// Grid: 4 row-tiles x 32 col-tiles = 128 waves.
// ---------------------------------------------------------------------------
__global__ void gru_kernel_k(const int* __restrict__ x,
                             const float* __restrict__ query,
                             const float* __restrict__ emb,
                             const float* __restrict__ dec_h,
                             const float* __restrict__ gk,   // [512,1536]
                             const float* __restrict__ grk,  // [512,1536]
                             const float* __restrict__ gb,   // [2,1536]
                             float* __restrict__ hnew,       // [64,512]
                             float* __restrict__ outcat,     // [64,1280]
                             float* __restrict__ state,      // [64,512]
                             float* __restrict__ output_)    // [64,512]
{
  const int wid  = blockIdx.x * 4 + (threadIdx.x >> 5);
  const int lane = threadIdx.x & 31;
  const int rowTile = wid >> 5;        // 0..3
  const int colTile = wid & 31;        // 0..31
  const int row0 = rowTile << 4;
  const int col0 = colTile << 4;
  const int m  = lane & 15;
  const int kh = (lane >> 4) << 1;     // 0 or 2

  const int arow = row0 + m;           // 0..63
  const int xi   = x[arow];

  const float* erow = emb   + (size_t)xi   * EMB_;
  const float* qrow = query + (size_t)arow * QD_;
  const float* hrow = dec_h + (size_t)arow * U_;

  v8f az = {}, ar = {}, ah = {};
  v8f bz = {}, br = {}, bh = {};

  for (int k = 0; k < 512; k += 4) {           // k is wave-uniform
    const float* ap = (k < 256) ? (erow + k) : (qrow + (k - 256));
    v2f a = *(const v2f*)(ap + kh);
    v2f h = *(const v2f*)(hrow + k + kh);

    const float* w0 = gk  + (size_t)(k + kh) * 1536 + col0 + m;
    const float* r0 = grk + (size_t)(k + kh) * 1536 + col0 + m;

    v2f b;
    b.x = w0[0];    b.y = w0[1536];        az = wmma4(a, b, az);
    b.x = w0[512];  b.y = w0[1536 + 512];  ar = wmma4(a, b, ar);
    b.x = w0[1024]; b.y = w0[1536 + 1024]; ah = wmma4(a, b, ah);
    b.x = r0[0];    b.y = r0[1536];        bz = wmma4(h, b, bz);
    b.x = r0[512];  b.y = r0[1536 + 512];  br = wmma4(h, b, br);
    b.x = r0[1024]; b.y = r0[1536 + 1024]; bh = wmma4(h, b, bh);
  }

  const int rbase = row0 + ((lane >> 4) << 3);
  const int c = col0 + m;
  for (int v = 0; v < 8; ++v) {
    const int r = rbase + v;
    float zz = fsigmoid(az[v] + gb[c]        + bz[v] + gb[1536 + c]);
    float rr = fsigmoid(ar[v] + gb[512 + c]  + br[v] + gb[2048 + c]);
    float hc = ftanh_hw(ah[v] + gb[1024 + c] + rr * (bh[v] + gb[2560 + c]));
    float ho = dec_h[(size_t)r * U_ + c];
    float hn = zz * ho + (1.0f - zz) * hc;
    hnew   [(size_t)r * U_ + c]          = hn;
    outcat [(size_t)r * 1280 + 256 + c]  = hn;
    state  [(size_t)r * U_ + c]          = hn;
    output_[(size_t)r * U_ + c]          = hn;
  }
}

// ---------------------------------------------------------------------------
// Generic f32 WMMA GEMM: C[MxN] = A[MxK] * B[KxN] + bias, one 16x16 tile/wave,
// row/col clamped for ragged edges (5000 = 312*16 + 8).
// ---------------------------------------------------------------------------
__global__ void gemm_tile_kernel(const float* __restrict__ A, int lda,
                                 const float* __restrict__ Bm, int ldb,
                                 const float* __restrict__ bias,
                                 float* __restrict__ C, int ldc,
                                 int K, int tilesM, int tilesN,
                                 int Mreal, int Nreal, int Mstore)
{
  const int wid = (blockIdx.x * blockDim.x + threadIdx.x) >> 5;
  if (wid >= tilesM * tilesN) return;            // wave-uniform exit
  const int lane = threadIdx.x & 31;
  const int rowTile = wid / tilesN;
  const int colTile = wid - rowTile * tilesN;
  const int row0 = rowTile << 4;
  const int col0 = colTile << 4;
  const int m  = lane & 15;
  const int kh = (lane >> 4) << 1;

  const int arow = imin(row0 + m, Mreal - 1);
  const int cn   = imin(col0 + m, Nreal - 1);
  const float* ap = A  + (size_t)arow * lda + kh;
  const float* bp = Bm + (size_t)kh   * ldb + cn;

  v8f acc = {};
  for (int k = 0; k < K; k += 4) {
    v2f a = *(const v2f*)(ap + k);
    v2f b;
    b.x = bp[(size_t)k * ldb];
    b.y = bp[(size_t)(k + 1) * ldb];
    acc = wmma4(a, b, acc);
  }

  const int rbase = row0 + ((lane >> 4) << 3);
  const int c = col0 + m;
  for (int v = 0; v < 8; ++v) {
    const int r = rbase + v;
    if (r < Mstore && c < Nreal)
      C[(size_t)r * ldc + c] = acc[v] + bias[c];
  }
}

// ---------------------------------------------------------------------------
// Attention: one block per batch row. score[p] = sum_u tanh(vproj[p,u]+q[u])*Vw[u]
// -> softmax over POI -> context[e] = sum_p attn[p]*emb[p,e].
// Scores staged in LDS (20 KB of 320 KB WGP LDS). 512 threads (16 waves).
// Also copies cat_dec_hidden into out_cat[:,768:1280].
// ---------------------------------------------------------------------------
__global__ void attn_kernel(const float* __restrict__ vproj,  // [5008,512]
                            const float* __restrict__ qproj,  // [64,512]
                            const float* __restrict__ Vw,     // [512]
                            const float* __restrict__ Vb,     // [1]
                            const float* __restrict__ emb,    // [5000,256]
                            const float* __restrict__ cat_dec,// [64,512]
                            float* __restrict__ outcat)       // [64,1280]
{
  __shared__ float q_s[U_];
  __shared__ float vw_s[U_];
  __shared__ float score_s[POI_];
  __shared__ float red_s[20];
  __shared__ float ctx_s[EMB_];

  const int b   = blockIdx.x;
  const int tid = threadIdx.x;        // 0..511

  outcat[(size_t)b * 1280 + 768 + tid] = cat_dec[(size_t)b * U_ + tid];
  q_s[tid]  = qproj[(size_t)b * U_ + tid];
  vw_s[tid] = Vw[tid];
  __syncthreads();

  const float vb = Vb[0];
  float lmax = -3.4e38f;
  for (int p = tid; p < POI_; p += 512) {
    const float* vp = vproj + (size_t)p * U_;
    float s = 0.f;
    for (int u = 0; u < U_; u += 4) {
      float4 vv = *(const float4*)(vp + u);
      s += ftanh_hw(vv.x + q_s[u])     * vw_s[u];
      s += ftanh_hw(vv.y + q_s[u + 1]) * vw_s[u + 1];
      s += ftanh_hw(vv.z + q_s[u + 2]) * vw_s[u + 2];
      s += ftanh_hw(vv.w + q_s[u + 3]) * vw_s[u + 3];
    }
    s += vb;
    score_s[p] = s;
    lmax = fmaxf(lmax, s);
  }
  // block-wide max
  for (int off = 16; off; off >>= 1) lmax = fmaxf(lmax, __shfl_xor(lmax, off, 32));
  if ((tid & 31) == 0) red_s[tid >> 5] = lmax;
  __syncthreads();
  if (tid == 0) {
    float mm = red_s[0];
    for (int i = 1; i < 16; ++i) mm = fmaxf(mm, red_s[i]);
    red_s[16] = mm;
  }
  __syncthreads();
  const float gmax = red_s[16];

  float lsum = 0.f;
  for (int p = tid; p < POI_; p += 512) {
    float e = fexp(score_s[p] - gmax);
    score_s[p] = e;
    lsum += e;
  }
  for (int off = 16; off; off >>= 1) lsum += __shfl_xor(lsum, off, 32);
  if ((tid & 31) == 0) red_s[tid >> 5] = lsum;
  __syncthreads();
  if (tid == 0) {
    float ss = 0.f;
    for (int i = 0; i < 16; ++i) ss += red_s[i];
    red_s[16] = ss;
  }
  __syncthreads();
  const float inv = 1.0f / red_s[16];

  // context: 512 threads = 2 halves x 256 embedding dims
  const int e    = tid & 255;
  const int half = tid >> 8;
  float ctx = 0.f;
  for (int p = half * 2500; p < (half + 1) * 2500; ++p)
    ctx += score_s[p] * emb[(size_t)p * EMB_ + e];
  if (half == 1) ctx_s[e] = ctx;
  __syncthreads();
  if (half == 0)
    outcat[(size_t)b * 1280 + e] = (ctx + ctx_s[e]) * inv;
}

// ---------------------------------------------------------------------------
extern "C" void kernel_launch(void* const* d_in, const int* in_sizes, int n_in,
                              void* d_out, int out_size, void* d_ws, size_t ws_size,
                              hipStream_t stream)
{
  const int*   x       = (const int*)  d_in[0];
  const float* query   = (const float*)d_in[1];
  const float* emb     = (const float*)d_in[2];
  // d_in[3] = A_hat (unused by reference)
  const float* dec_h   = (const float*)d_in[4];
  const float* cat_dec = (const float*)d_in[5];
  const float* gk      = (const float*)d_in[6];
  const float* grk     = (const float*)d_in[7];
  const float* gb      = (const float*)d_in[8];
  const float* W1w     = (const float*)d_in[9];
  const float* W1b     = (const float*)d_in[10];
  const float* W2w     = (const float*)d_in[11];
  const float* W2b     = (const float*)d_in[12];
  const float* Vw      = (const float*)d_in[13];
  const float* Vb      = (const float*)d_in[14];
  const float* fcw     = (const float*)d_in[15];
  const float* fcb     = (const float*)d_in[16];

  float* out     = (float*)d_out;
  float* logits  = out;                       // [64,5000]
  float* state   = out + B_ * POI_;           // [64,512]
  float* output_ = state + B_ * U_;           // [64,512]

  float* ws     = (float*)d_ws;               // ~10.4 MB used
  float* vproj  = ws;                         // [5008,512]
  float* qproj  = vproj + 5008 * U_;          // [64,512]
  float* hnew   = qproj + B_ * U_;            // [64,512]
  float* outcat = hnew  + B_ * U_;            // [64,1280]

  // 1) GRU (fills h_new, out_cat[:,256:768], state, output_)
  gru_kernel_k<<<32, 128, 0, stream>>>(x, query, emb, dec_h, gk, grk, gb,
                                       hnew, outcat, state, output_);

  // 2) v_proj = emb @ W1 + b1 : [5000,256]x[256,512], tiles 313x32
  gemm_tile_kernel<<<(313 * 32 + 7) / 8, 256, 0, stream>>>(
      emb, EMB_, W1w, U_, W1b, vproj, U_,
      /*K=*/EMB_, /*tilesM=*/313, /*tilesN=*/32,
      /*Mreal=*/POI_, /*Nreal=*/U_, /*Mstore=*/5008);

  // 3) q_proj = h_new @ W2 + b2 : [64,512]x[512,512], tiles 4x32
  gemm_tile_kernel<<<(4 * 32 + 7) / 8, 256, 0, stream>>>(
      hnew, U_, W2w, U_, W2b, qproj, U_,
      /*K=*/U_, /*tilesM=*/4, /*tilesN=*/32,
      /*Mreal=*/B_, /*Nreal=*/U_, /*Mstore=*/B_);

  // 4) score/softmax/context (also copies cat_dec_hidden into out_cat)
  attn_kernel<<<B_, 512, 0, stream>>>(vproj, qproj, Vw, Vb, emb, cat_dec, outcat);

  // 5) logits = out_cat @ fc_w + fc_b : [64,1280]x[1280,5000], tiles 4x313
  gemm_tile_kernel<<<(4 * 313 + 7) / 8, 256, 0, stream>>>(
      outcat, 1280, fcw, POI_, fcb, logits, POI_,
      /*K=*/1280, /*tilesM=*/4, /*tilesN=*/313,
      /*Mreal=*/B_, /*Nreal=*/POI_, /*Mstore=*/B_);
}